// TaskAlignedAssigner_11553462026342
// MI455X (gfx1250) — compile-verified
//
#include <hip/hip_runtime.h>

#define BS   16
#define NMAX 64
#define NA   8400
#define NC   80
#define TOPKK 13
#define FEPS 1e-7f
#define GEPS 1e-9f

typedef __attribute__((ext_vector_type(2))) float v2f;
typedef __attribute__((ext_vector_type(8))) float v8f;

// -------------------------------------------------------------------------
// K1: bbox_scores[b,n,a] = pd_scores[b,a,label[b,n]] as an EXACT fp32 WMMA
// GEMM:  S = onehot(labels)[64x80] x P^T[80x8400]   (per batch)
// One wave computes one 16(M=gt) x 16(N=anchor) tile, K=80 in steps of 4.
// A-frag (16x4 f32): lanes 0-15 = M rows, VGPR0/1 = K{0,1}; lanes 16-31 K{2,3}
// B-frag (4x16 f32): lanes 0-15 = N cols, VGPR0/1 = K{0,1}; lanes 16-31 K{2,3}
// C/D (16x16 f32):  acc[r]: lanes 0-15 -> M=r, lanes 16-31 -> M=r+8
// -------------------------------------------------------------------------
__global__ void k_scores_wmma(const float* __restrict__ pd,
                              const int* __restrict__ lblp,
                              float* __restrict__ sbuf) {
  const int wave = (blockIdx.x * blockDim.x + threadIdx.x) >> 5;
  const int lane = threadIdx.x & 31;
  const int nt = wave % (NA / 16);
  const int t  = wave / (NA / 16);
  const int mt = t & 3;          // 4 M-tiles (64 GTs)
  const int b  = t >> 2;         // batch
  const int half = lane >> 4;    // 0: K{0,1}, 1: K{2,3}
  const int l    = lane & 15;
  const int m0 = mt * 16, a0 = nt * 16;
  const int lbl = lblp[b * NMAX + m0 + l];       // label of GT row m0+l
  const float* Pb = pd + ((size_t)b * NA + (size_t)(a0 + l)) * NC;

  v8f acc = {0.f, 0.f, 0.f, 0.f, 0.f, 0.f, 0.f, 0.f};
#pragma unroll
  for (int kt = 0; kt < NC / 4; ++kt) {
    const int kb = kt * 4 + half * 2;
    v2f af;                                      // exact one-hot A fragment
    af.x = (lbl == kb)     ? 1.f : 0.f;
    af.y = (lbl == kb + 1) ? 1.f : 0.f;
    v2f bf = *(const v2f*)(Pb + kb);             // B = P^T tile, coalesced
    acc = __builtin_amdgcn_wmma_f32_16x16x4_f32(
        false, af, false, bf, (short)0, acc, false, false);
  }
  float* o = sbuf + (size_t)b * NMAX * NA + a0 + l;
  const int mb = m0 + half * 8;
#pragma unroll
  for (int r = 0; r < 8; ++r) o[(size_t)(mb + r) * NA] = acc[r];
}

// -------------------------------------------------------------------------
// K2: one block per (b,n) row. Pass 1: CIoU overlaps (stored) + row maxima
// of the 3 cost components (LDS reduce). Pass 2: normalized cost, fuse
// align_metric = score * overlap^6 * cost.
// -------------------------------------------------------------------------
__global__ void k_pair(const float* __restrict__ pdb, const float* __restrict__ gtb,
                       const float* __restrict__ sbuf, float* __restrict__ ovbuf,
                       float* __restrict__ ambuf) {
  const int row = blockIdx.x;             // b*NMAX + n
  const int b = row / NMAX;
  const int tid = threadIdx.x;
  const float g0 = gtb[row*4+0], g1 = gtb[row*4+1];
  const float g2 = gtb[row*4+2], g3 = gtb[row*4+3];
  const float w1 = g2 - g0, h1 = (g3 - g1) + FEPS;
  const float atan1 = atanf(w1 / h1);
  const float gtr = fminf(w1, h1) / fmaxf(w1, h1);
  const float ga  = w1 * h1;
  const float r1  = w1 / h1;
  const float C4PI2 = 4.f / (3.14159265358979323846f * 3.14159265358979323846f);

  __shared__ float rs[256], ra[256], rr[256];
  float ms = -1e30f, ma = -1e30f, mr = -1e30f;
  const float* P = pdb + (size_t)b * NA * 4;
  float* OV = ovbuf + (size_t)row * NA;

  for (int a = tid; a < NA; a += 256) {
    const float p0 = P[a*4+0], p1 = P[a*4+1], p2 = P[a*4+2], p3 = P[a*4+3];
    const float w2 = p2 - p0, h2 = (p3 - p1) + FEPS;
    float iw = fmaxf(fminf(g2, p2) - fmaxf(g0, p0), 0.f);
    float ih = fmaxf(fminf(g3, p3) - fmaxf(g1, p1), 0.f);
    float inter = iw * ih;
    float uni = w1 * h1 + w2 * h2 - inter + FEPS;
    float iou = inter / uni;
    float cw = fmaxf(g2, p2) - fminf(g0, p0);
    float ch = fmaxf(g3, p3) - fminf(g1, p1);
    float c2 = cw * cw + ch * ch + FEPS;
    float dx = (p0 + p2 - g0 - g2), dy = (p1 + p3 - g1 - g3);
    float rho2 = (dx * dx + dy * dy) * 0.25f;
    float dat = atanf(w2 / h2) - atan1;
    float v = C4PI2 * dat * dat;
    float alpha = v / (v - iou + (1.f + FEPS));
    OV[a] = fmaxf(iou - (rho2 / c2 + v * alpha), 0.f);
    float pdr = fminf(w2, h2) / fmaxf(w2, h2);
    float cs  = 1.f - fabsf(gtr - pdr) / fmaxf(gtr, pdr);
    float pa  = w2 * h2;
    float ca  = 1.f - fabsf(ga - pa) / fmaxf(ga, pa);
    float r2  = w2 / h2;
    float cwh = fminf(r1, r2) / fmaxf(r1, r2);
    ms = fmaxf(ms, cs); ma = fmaxf(ma, ca); mr = fmaxf(mr, cwh);
  }
  rs[tid] = ms; ra[tid] = ma; rr[tid] = mr; __syncthreads();
  for (int s = 128; s > 0; s >>= 1) {
    if (tid < s) {
      rs[tid] = fmaxf(rs[tid], rs[tid+s]);
      ra[tid] = fmaxf(ra[tid], ra[tid+s]);
      rr[tid] = fmaxf(rr[tid], rr[tid+s]);
    }
    __syncthreads();
  }
  const float ds = rs[0] + FEPS, da = ra[0] + FEPS, dr = rr[0] + FEPS;
  const float* SC = sbuf + (size_t)row * NA;
  float* AM = ambuf + (size_t)row * NA;
  for (int a = tid; a < NA; a += 256) {
    const float p0 = P[a*4+0], p1 = P[a*4+1], p2 = P[a*4+2], p3 = P[a*4+3];
    const float w2 = p2 - p0, h2 = (p3 - p1) + FEPS;
    float pdr = fminf(w2, h2) / fmaxf(w2, h2);
    float cs  = 1.f - fabsf(gtr - pdr) / fmaxf(gtr, pdr);
    float pa  = w2 * h2;
    float ca  = 1.f - fabsf(ga - pa) / fmaxf(ga, pa);
    float r2  = w2 / h2;
    float cwh = fminf(r1, r2) / fmaxf(r1, r2);
    float cost = fmaxf((cs / ds + cwh / dr + ca / da) * (1.f / 3.f), 0.f);
    float ov = OV[a];
    float ov2 = ov * ov;
    AM[a] = SC[a] * (ov2 * ov2 * ov2) * cost;
  }
}

// -------------------------------------------------------------------------
// K3: top-13 per (b,n) row. Masked metric staged in LDS; 13 argmax rounds
// with lowest-index tie-break (matches lax.top_k set); each mask_pos element
// written exactly once at the end.
// -------------------------------------------------------------------------
__global__ void k_topk(const float* __restrict__ gtb, const float* __restrict__ anc,
                       const float* __restrict__ mgt, const float* __restrict__ ambuf,
                       float* __restrict__ mpbuf) {
  const int row = blockIdx.x;
  const int tid = threadIdx.x;
  __shared__ float smet[NA];
  __shared__ unsigned char sing[NA];
  __shared__ float redv[256];
  __shared__ int   redi[256];
  __shared__ int   sel[TOPKK];
  const float lx = gtb[row*4+0], ly = gtb[row*4+1];
  const float rx = gtb[row*4+2], ry = gtb[row*4+3];
  const float mg = mgt[row];
  const float* AM = ambuf + (size_t)row * NA;
  float* MP = mpbuf + (size_t)row * NA;

  for (int a = tid; a < NA; a += 256) {
    float ax = anc[a*2+0], ay = anc[a*2+1];
    float mn = fminf(fminf(ax - lx, ay - ly), fminf(rx - ax, ry - ay));
    bool ing = (mn > GEPS);
    sing[a] = ing ? 1 : 0;
    smet[a] = ing ? AM[a] : 0.f;     // metrics are >= 0
  }
  __syncthreads();
  if (mg > 0.f) {                    // uniform per block
    for (int t = 0; t < TOPKK; ++t) {
      float bv = -1.f; int bi = 0x7fffffff;
      for (int a = tid; a < NA; a += 256) {
        float v = smet[a];
        if (v > bv) { bv = v; bi = a; }   // strict > keeps lowest index
      }
      redv[tid] = bv; redi[tid] = bi; __syncthreads();
      for (int s = 128; s > 0; s >>= 1) {
        if (tid < s) {
          if (redv[tid+s] > redv[tid] ||
              (redv[tid+s] == redv[tid] && redi[tid+s] < redi[tid])) {
            redv[tid] = redv[tid+s]; redi[tid] = redi[tid+s];
          }
        }
        __syncthreads();
      }
      if (tid == 0) { sel[t] = redi[0]; smet[redi[0]] = -1.f; }
      __syncthreads();
    }
  }
  // single write per element: selected & in-gt -> 1, else 0
  for (int a = tid; a < NA; a += 256) {
    float v = 0.f;
    if (mg > 0.f) {
#pragma unroll
      for (int t = 0; t < TOPKK; ++t)
        if (sel[t] == a) v = sing[a] ? 1.f : 0.f;
    }
    MP[a] = v;
  }
}

// -------------------------------------------------------------------------
// K4: per (b,a): fg count; if >1 rewrite column to one-hot of argmax overlap;
// target_gt_idx, fg_mask, gather labels/bboxes.
// -------------------------------------------------------------------------
__global__ void k_resolve(const float* __restrict__ ovbuf, float* __restrict__ mpbuf,
                          const int* __restrict__ lblp, const float* __restrict__ gtb,
                          int* __restrict__ tgi, float* __restrict__ out_lab,
                          float* __restrict__ out_box, float* __restrict__ out_fg) {
  const int q = blockIdx.x * 256 + threadIdx.x;   // b*NA + a
  if (q >= BS * NA) return;
  const int b = q / NA, a = q % NA;
  const size_t base = (size_t)b * NMAX * NA + a;
  float s = 0.f;
  for (int n = 0; n < NMAX; ++n) s += mpbuf[base + (size_t)n * NA];
  int gi = 0; float fg;
  if (s > 1.5f) {                                 // multi-GT anchor
    float bv = -1e30f; int bi = 0;
    for (int n = 0; n < NMAX; ++n) {
      float v = ovbuf[base + (size_t)n * NA];
      if (v > bv) { bv = v; bi = n; }
    }
    for (int n = 0; n < NMAX; ++n)
      mpbuf[base + (size_t)n * NA] = (n == bi) ? 1.f : 0.f;
    gi = bi; fg = 1.f;
  } else {
    for (int n = 0; n < NMAX; ++n)
      if (mpbuf[base + (size_t)n * NA] > 0.5f) { gi = n; break; }
    fg = (s > 0.f) ? 1.f : 0.f;
  }
  tgi[q] = gi;
  out_fg[q] = fg;
  out_lab[q] = (float)lblp[b * NMAX + gi];
  const float* gb = gtb + (size_t)(b * NMAX + gi) * 4;
  out_box[(size_t)q*4+0] = gb[0]; out_box[(size_t)q*4+1] = gb[1];
  out_box[(size_t)q*4+2] = gb[2]; out_box[(size_t)q*4+3] = gb[3];
}

// K5: per row: ratio = max(ov*mp) / (max(am*mp) + EPS)
__global__ void k_rowstat(const float* __restrict__ ambuf, const float* __restrict__ ovbuf,
                          const float* __restrict__ mpbuf, float* __restrict__ ratio) {
  const int row = blockIdx.x, tid = threadIdx.x;
  __shared__ float r1s[256], r2s[256];
  float pa = 0.f, po = 0.f;            // all candidates >= 0
  const size_t o = (size_t)row * NA;
  for (int a = tid; a < NA; a += 256) {
    float m = mpbuf[o + a];
    pa = fmaxf(pa, ambuf[o + a] * m);
    po = fmaxf(po, ovbuf[o + a] * m);
  }
  r1s[tid] = pa; r2s[tid] = po; __syncthreads();
  for (int s = 128; s > 0; s >>= 1) {
    if (tid < s) { r1s[tid] = fmaxf(r1s[tid], r1s[tid+s]);
                   r2s[tid] = fmaxf(r2s[tid], r2s[tid+s]); }
    __syncthreads();
  }
  if (tid == 0) ratio[row] = r2s[0] / (r1s[0] + GEPS);
}

// K6a: norm[b,a] = max_n (am*mp*ratio)
__global__ void k_norm(const float* __restrict__ ambuf, const float* __restrict__ mpbuf,
                       const float* __restrict__ ratio, float* __restrict__ nrm) {
  const int q = blockIdx.x * 256 + threadIdx.x;
  if (q >= BS * NA) return;
  const int b = q / NA, a = q % NA;
  float nm = 0.f;
  for (int n = 0; n < NMAX; ++n) {
    const int row = b * NMAX + n;
    const size_t o = (size_t)row * NA + a;
    nm = fmaxf(nm, ambuf[o] * mpbuf[o] * ratio[row]);
  }
  nrm[q] = nm;
}

// K6b: target_scores = onehot(label) * fg * norm (coalesced over class dim)
__global__ void k_scoresout(const int* __restrict__ lblp, const int* __restrict__ tgi,
                            const float* __restrict__ nrm, const float* __restrict__ out_fg,
                            float* __restrict__ out_sc) {
  const long long t = (long long)blockIdx.x * 256 + threadIdx.x;
  if (t >= (long long)BS * NA * NC) return;
  const int c = (int)(t % NC);
  const int q = (int)(t / NC);
  const int b = q / NA;
  const int lbl = lblp[b * NMAX + tgi[q]];
  out_sc[(size_t)q * NC + c] = (out_fg[q] > 0.f && c == lbl) ? nrm[q] : 0.f;
}

extern "C" void kernel_launch(void* const* d_in, const int* in_sizes, int n_in,
                              void* d_out, int out_size, void* d_ws, size_t ws_size,
                              hipStream_t stream) {
  (void)in_sizes; (void)n_in; (void)out_size; (void)ws_size;
  const float* pd_scores = (const float*)d_in[0];
  const float* pd_bboxes = (const float*)d_in[1];
  const float* anc       = (const float*)d_in[2];
  const int*   gt_labels = (const int*)d_in[3];
  const float* gt_bboxes = (const float*)d_in[4];
  const float* mask_gt   = (const float*)d_in[5];

  float* ws = (float*)d_ws;
  const size_t PAIR = (size_t)BS * NMAX * NA;       // 8,601,600
  float* ovbuf = ws;                                 // overlaps
  float* ambuf = ws + PAIR;                          // align_metric
  float* spbuf = ws + 2 * PAIR;                      // bbox_scores, then mask_pos
  float* ratio = ws + 3 * PAIR;                      // B*NMAX
  int*   tgi   = (int*)(ws + 3 * PAIR + BS * NMAX); // B*NA
  float* nrm   = ws + 3 * PAIR + BS * NMAX + (size_t)BS * NA;

  float* out_lab = (float*)d_out;
  float* out_box = out_lab + (size_t)BS * NA;
  float* out_sc  = out_box + (size_t)BS * NA * 4;
  float* out_fg  = out_sc  + (size_t)BS * NA * NC;

  k_scores_wmma<<<(BS * 4 * (NA / 16)) / 8, 256, 0, stream>>>(pd_scores, gt_labels, spbuf);
  k_pair     <<<BS * NMAX, 256, 0, stream>>>(pd_bboxes, gt_bboxes, spbuf, ovbuf, ambuf);
  k_topk     <<<BS * NMAX, 256, 0, stream>>>(gt_bboxes, anc, mask_gt, ambuf, spbuf);
  k_resolve  <<<(BS * NA) / 256, 256, 0, stream>>>(ovbuf, spbuf, gt_labels, gt_bboxes,
                                                   tgi, out_lab, out_box, out_fg);
  k_rowstat  <<<BS * NMAX, 256, 0, stream>>>(ambuf, ovbuf, spbuf, ratio);
  k_norm     <<<(BS * NA) / 256, 256, 0, stream>>>(ambuf, spbuf, ratio, nrm);
  k_scoresout<<<((long long)BS * NA * NC) / 256, 256, 0, stream>>>(gt_labels, tgi, nrm,
                                                                   out_fg, out_sc);
}